// Integrated_Co_GA_Loss_8564164788761
// MI455X (gfx1250) — compile-verified
//
#include <hip/hip_runtime.h>
#include <hip/hip_bf16.h>

typedef __attribute__((ext_vector_type(2))) float v2f;
typedef __attribute__((ext_vector_type(8))) float v8f;

// Wave32 reduction via one V_WMMA_F32_16X16X4_F32:
//   A (16x4): lane L, VGPR0 -> (M=L%16, K = L<16 ? 0 : 2), VGPR1 = 0
//   B (4x16): all ones  =>  D[m][n] = s_m + s_{m+16}  (row sums of A)
//   C/D layout: c[j] holds row j (lanes 0-15) and row j+8 (lanes 16-31)
//   sum_j c[j] gives rows 0-7 (low half) / 8-15 (high half); xor-16 combines.
__device__ __forceinline__ float wave32_sum_wmma(float s) {
    v2f a;    a.x = s;    a.y = 0.0f;
    v2f ones; ones.x = 1.0f; ones.y = 1.0f;
    v8f c = {};
    c = __builtin_amdgcn_wmma_f32_16x16x4_f32(false, a, false, ones,
                                              (short)0, c, false, false);
    float t = c[0] + c[1] + c[2] + c[3] + c[4] + c[5] + c[6] + c[7];
    t += __shfl_xor(t, 16, 32);
    return t; // every lane holds the full 32-lane sum
}

// 256-thread block reduction; result valid in thread 0 only.
__device__ __forceinline__ float block256_sum(float s) {
    __shared__ float smem[8];
    const float w = wave32_sum_wmma(s);
    const int lane = threadIdx.x & 31;
    const int wid  = threadIdx.x >> 5;
    if (lane == 0) smem[wid] = w;
    __syncthreads();
    float total = 0.0f;
    if (threadIdx.x == 0) {
#pragma unroll
        for (int i = 0; i < 8; ++i) total += smem[i];
    }
    return total;
}

__global__ __launch_bounds__(256)
void coga_loss_main(const float* __restrict__ inputs,
                    const int*   __restrict__ targets,
                    int nrows, float* __restrict__ partials) {
    const int tid    = blockIdx.x * blockDim.x + threadIdx.x;
    const int stride = gridDim.x * blockDim.x;
    const float4* __restrict__ in4 = (const float4*)inputs;

    float acc = 0.0f;
    for (int row = tid; row < nrows; row += stride) {
        if (row + stride < nrows)
            __builtin_prefetch(&in4[(size_t)(row + stride) * 2], 0, 0);

        const float4 lo = in4[(size_t)row * 2];       // global_load_b128
        const float4 hi = in4[(size_t)row * 2 + 1];   // global_load_b128
        float x[8] = {lo.x, lo.y, lo.z, lo.w, hi.x, hi.y, hi.z, hi.w};
        const int t = targets[row] & 7;

        // top-2 of the logits (monotone under softmax)
        float m1 = x[0], m2 = -3.4e38f;
#pragma unroll
        for (int i = 1; i < 8; ++i) {
            const float v = x[i];
            if (v > m1)      { m2 = m1; m1 = v; }
            else if (v > m2) { m2 = v; }
        }

        float se = 0.0f;
#pragma unroll
        for (int i = 0; i < 8; ++i) se += __expf(x[i] - m1);
        const float lse = m1 + __logf(se);

        const float xt   = x[t];
        const float ce   = lse - xt;           // -log_softmax[t]
        const float pt   = __expf(xt - lse);
        const float pmax = __expf(m1 - lse);
        const float p2   = __expf(m2 - lse);

        const bool  ispt = (pmax == pt);
        const float pneg = ispt ? p2 : pmax;

        // q = d^T * inv_cov * d ; inv(T_COV)=[[2.4,1.6],[1.6,2.4]]
        //                          inv(F_COV)=[[10.30303,-9.69697],[-9.69697,10.30303]]
        float dx, dy, q;
        if (ispt) {
            dx = pt - 0.5f;  dy = pneg - 0.5f;
            q  = 2.4f * (dx * dx + dy * dy) + 3.2f * (dx * dy);
        } else {
            dx = pt - 0.3f;  dy = pneg - 0.15f;
            q  = 10.3030303f * (dx * dx + dy * dy) - 19.3939394f * (dx * dy);
        }
        acc += ce * __expf(-0.5f * q);
    }

    const float bsum = block256_sum(acc);
    if (threadIdx.x == 0) partials[blockIdx.x] = bsum;
}

__global__ __launch_bounds__(256)
void coga_loss_finish(const float* __restrict__ partials, int n, float inv_n,
                      float* __restrict__ out) {
    float s = 0.0f;
    for (int i = threadIdx.x; i < n; i += 256) s += partials[i];
    const float total = block256_sum(s);
    if (threadIdx.x == 0) out[0] = total * inv_n;
}

extern "C" void kernel_launch(void* const* d_in, const int* in_sizes, int n_in,
                              void* d_out, int out_size, void* d_ws, size_t ws_size,
                              hipStream_t stream) {
    const float* inputs  = (const float*)d_in[0];
    const int*   targets = (const int*)d_in[1];
    const int    nrows   = in_sizes[1];            // N (targets count)

    // ~8 rows per thread at 2048 blocks x 256 threads for N = 4M.
    int nblocks = 2048;
    const int cap = (int)(ws_size / sizeof(float));
    if (cap > 0 && nblocks > cap) nblocks = cap;
    if (nblocks < 1) nblocks = 1;

    float* partials = (float*)d_ws;
    coga_loss_main<<<nblocks, 256, 0, stream>>>(inputs, targets, nrows, partials);
    coga_loss_finish<<<1, 256, 0, stream>>>(partials, nblocks,
                                            1.0f / (float)nrows, (float*)d_out);
}